// Projection_73478300500512
// MI455X (gfx1250) — compile-verified
//
#include <hip/hip_runtime.h>

// Problem constants (match reference): N=2048, C=4, M=64, D=128
#define PN 2048
#define PC 4
#define PM 64
#define PD 128
#define NT 14          // Horner polynomial order (||scaled A||_1 <= 1 -> ~1/14! error)
#define MAXSQ 16       // reference MAX_S

typedef float v2f __attribute__((ext_vector_type(2)));
typedef float v8f __attribute__((ext_vector_type(8)));

// XOR swizzle for the H plane: conflict-free column (A-operand) and row (B-operand) reads.
__device__ __forceinline__ int HIDX(int r, int c) { return r * 64 + (c ^ (r & 31)); }

__device__ __forceinline__ v8f wmma_f32(v2f a, v2f b, v8f c) {
  // V_WMMA_F32_16X16X4_F32 : D(16x16 f32) = A(16x4 f32) x B(4x16 f32) + C
  return __builtin_amdgcn_wmma_f32_16x16x4_f32(false, a, false, b, (short)0, c,
                                               false, false);
}

// ---------------------------------------------------------------------------
// Kernel 1: AdX[n,c,h,i] = sum_d 0.5*(W[c,h,i,d] - conj(W[c,i,h,d])) * dX[n,d]
// One wave computes a 16(n) x 16(hi) complex tile. Output interleaved (re,im)
// at d_out[(n*C + c)*4096 + h*64 + i]  (== complex64 layout).
// ---------------------------------------------------------------------------
__global__ __launch_bounds__(256) void adx_gemm_kernel(
    const float* __restrict__ dX, const float* __restrict__ Wre,
    const float* __restrict__ Wim, float2* __restrict__ out) {
  const int tid  = threadIdx.x;
  const int lane = tid & 31;
  const int wv   = tid >> 5;                    // 8 waves: 2 (n) x 4 (hi)
  const int c    = blockIdx.z;
  const int nBase  = blockIdx.x * 32 + (wv >> 2) * 16;
  const int hiBase = blockIdx.y * 64 + (wv & 3) * 16;

  const int m  = lane & 15;
  const int kh = (lane >> 4) << 1;              // k offset for this half-wave

  // B operand: column hi is fixed per lane across all K-steps
  const int hi = hiBase + m;
  const int h  = hi >> 6;
  const int i  = hi & 63;
  const float* pReHI = Wre + ((long)((c * 64 + h) * 64 + i)) * PD;
  const float* pReIH = Wre + ((long)((c * 64 + i) * 64 + h)) * PD;
  const float* pImHI = Wim + ((long)((c * 64 + h) * 64 + i)) * PD;
  const float* pImIH = Wim + ((long)((c * 64 + i) * 64 + h)) * PD;
  // A operand: row n fixed per lane
  const float* pX = dX + (long)(nBase + m) * PD;

  v8f accr = {}, acci = {};
#pragma unroll
  for (int s = 0; s < PD / 4; ++s) {
    const int d0 = s * 4 + kh;                  // even -> 8B aligned float2 loads
    const float2 x   = *(const float2*)(pX + d0);
    const float2 w1r = *(const float2*)(pReHI + d0);
    const float2 w2r = *(const float2*)(pReIH + d0);
    const float2 w1i = *(const float2*)(pImHI + d0);
    const float2 w2i = *(const float2*)(pImIH + d0);
    v2f a  = {x.x, x.y};
    v2f br = {0.5f * (w1r.x - w2r.x), 0.5f * (w1r.y - w2r.y)};
    v2f bi = {0.5f * (w1i.x + w2i.x), 0.5f * (w1i.y + w2i.y)};
    accr = wmma_f32(a, br, accr);
    acci = wmma_f32(a, bi, acci);
  }

  const int rE = (lane >> 4) << 3;
  const int cE = lane & 15;
#pragma unroll
  for (int r = 0; r < 8; ++r) {
    const int n   = nBase + rE + r;
    const int hiO = hiBase + cE;                // contiguous across lanes
    float2 v; v.x = accr[r]; v.y = acci[r];
    out[((long)n * PC + c) * (PM * PM) + hiO] = v;
  }
}

// ---------------------------------------------------------------------------
// Kernel 2: per-block 64x64 complex expm via Horner Taylor + repeated squaring.
// S plane: linear interleaved complex; H plane: XOR-swizzled interleaved.
// Each wave computes TWO row-adjacent 16x16 tiles in the SAME column strip so
// the B fragments are shared: 6 ds_load_b64 per 8 WMMAs.
// ---------------------------------------------------------------------------
template <bool BSWZ>
__device__ __forceinline__ void cmm_dual(const float2* __restrict__ A,
                                         const float2* __restrict__ B,
                                         int rowBase, int colBase, int lane,
                                         v8f& acc0r, v8f& acc0i,
                                         v8f& acc1r, v8f& acc1i) {
  const int m    = lane & 15;
  const int kh   = (lane >> 4) << 1;
  const int row0 = rowBase + m;                 // A-operand rows (tile 0)
  const int row1 = rowBase + 16 + m;            // A-operand rows (tile 1)
  const int col  = colBase + m;                 // shared B-operand column
#pragma unroll
  for (int kk = 0; kk < 16; ++kk) {
    const int k0 = kk * 4 + kh;
    const float2 a00 = A[HIDX(row0, k0)];
    const float2 a01 = A[HIDX(row0, k0 + 1)];
    const float2 a10 = A[HIDX(row1, k0)];
    const float2 a11 = A[HIDX(row1, k0 + 1)];
    const float2 b0  = BSWZ ? B[HIDX(k0, col)]     : B[k0 * 64 + col];
    const float2 b1  = BSWZ ? B[HIDX(k0 + 1, col)] : B[(k0 + 1) * 64 + col];
    v2f br = {b0.x, b1.x}, bi = {b0.y, b1.y};
    v2f a0r = {a00.x, a01.x}, a0i = {a00.y, a01.y};
    v2f a1r = {a10.x, a11.x}, a1i = {a10.y, a11.y};
    v2f n0i = {-a0i.x, -a0i.y};                 // f32 WMMA has no A-neg modifier
    v2f n1i = {-a1i.x, -a1i.y};
    acc0r = wmma_f32(a0r, br, acc0r);
    acc0r = wmma_f32(n0i, bi, acc0r);
    acc0i = wmma_f32(a0r, bi, acc0i);
    acc0i = wmma_f32(a0i, br, acc0i);
    acc1r = wmma_f32(a1r, br, acc1r);
    acc1r = wmma_f32(n1i, bi, acc1r);
    acc1i = wmma_f32(a1r, bi, acc1i);
    acc1i = wmma_f32(a1i, br, acc1i);
  }
}

__global__ __launch_bounds__(256) void expm64_kernel(float2* __restrict__ io) {
  __shared__ float2 Sc[64 * 64];                // 32 KB, linear
  __shared__ float2 Hc[64 * 64];                // 32 KB, swizzled
  const int tid  = threadIdx.x;
  const int lane = tid & 31;
  const int wv   = tid >> 5;
  const long base = (long)blockIdx.x * (PM * PM);

  // Load AdX (linear idx == h*64+i) with async copy straight into LDS:
  // 256 threads x 8 iters x 16B = 32 KB. Tracked by ASYNCcnt.
  {
    const char* gsrc = (const char*)(io + base);
    const unsigned lbase = (unsigned)(unsigned long long)(&Sc[0]);
#pragma unroll
    for (int t = 0; t < 8; ++t) {
      const int idx4 = tid + t * 256;           // 16-byte chunk index
      unsigned long long gaddr = (unsigned long long)(gsrc + (long)idx4 * 16);
      unsigned laddr = lbase + (unsigned)idx4 * 16u;
      asm volatile("global_load_async_to_lds_b128 %0, %1, off"
                   :: "v"(laddr), "v"(gaddr) : "memory");
    }
    asm volatile("s_wait_asynccnt 0x0" ::: "memory");
  }
  __syncthreads();

  // 1-norm: max over columns i of sum over rows h of |AdX|
  float* scratch = (float*)Hc;                  // H not initialized yet
  if (tid < 64) {
    float s = 0.f;
    for (int r = 0; r < 64; ++r) {
      const float2 v = Sc[r * 64 + tid];
      s += sqrtf(v.x * v.x + v.y * v.y);
    }
    scratch[tid] = s;
  }
  __syncthreads();
  if (tid == 0) {
    float mx = 0.f;
    for (int j = 0; j < 64; ++j) mx = fmaxf(mx, scratch[j]);
    const float sf = (mx > 1.f) ? ceilf(log2f(mx)) : 0.f;
    scratch[64] = exp2f(-sf);
    int si = (int)sf;
    ((int*)scratch)[65] = (si < MAXSQ) ? si : MAXSQ;
  }
  __syncthreads();
  const float scl = scratch[64];
  const int   nsq = ((int*)scratch)[65];
  __syncthreads();                              // done reading scratch before H reuse

  // S *= 2^-s ; H = I + S/NT
  const float invNT = 1.0f / (float)NT;
#pragma unroll
  for (int t = 0; t < 16; ++t) {
    const int idx = tid + t * 256;
    const int r = idx >> 6, cc = idx & 63;
    float2 v = Sc[idx];
    v.x *= scl; v.y *= scl;
    Sc[idx] = v;
    float2 hcv;
    hcv.x = v.x * invNT + (r == cc ? 1.f : 0.f);
    hcv.y = v.y * invNT;
    Hc[HIDX(r, cc)] = hcv;
  }
  __syncthreads();

  // Tile assignment: wave -> one 16-wide column strip, two row-adjacent tiles.
  const int cb = (wv & 3) * 16;
  const int rb = (wv >> 2) * 32;                // tiles at rb and rb+16
  const int rE = (lane >> 4) << 3;
  const int cE = lane & 15;

  // Horner: H <- I + (H*S)/k   (H*S == S*H: both are polynomials in S)
#pragma unroll 1
  for (int k = NT - 1; k >= 1; --k) {
    const float inv = 1.0f / (float)k;
    v8f p0r = {}, p0i = {}, p1r = {}, p1i = {};
    cmm_dual<false>(Hc, Sc, rb, cb, lane, p0r, p0i, p1r, p1i);
    __syncthreads();                            // all reads of H complete
#pragma unroll
    for (int r = 0; r < 8; ++r) {
      const int r0 = rb + rE + r, c0 = cb + cE;
      float2 hv; hv.x = p0r[r] * inv + (r0 == c0 ? 1.f : 0.f); hv.y = p0i[r] * inv;
      Hc[HIDX(r0, c0)] = hv;
      const int r1 = rb + 16 + rE + r;
      float2 hw; hw.x = p1r[r] * inv + (r1 == c0 ? 1.f : 0.f); hw.y = p1i[r] * inv;
      Hc[HIDX(r1, c0)] = hw;
    }
    __syncthreads();
  }

  // Repeated squaring: H <- H*H, nsq times (block-uniform trip count)
#pragma unroll 1
  for (int q = 0; q < nsq; ++q) {
    v8f p0r = {}, p0i = {}, p1r = {}, p1i = {};
    cmm_dual<true>(Hc, Hc, rb, cb, lane, p0r, p0i, p1r, p1i);
    __syncthreads();
#pragma unroll
    for (int r = 0; r < 8; ++r) {
      const int c0 = cb + cE;
      float2 hv; hv.x = p0r[r]; hv.y = p0i[r];
      Hc[HIDX(rb + rE + r, c0)] = hv;
      float2 hw; hw.x = p1r[r]; hw.y = p1i[r];
      Hc[HIDX(rb + 16 + rE + r, c0)] = hw;
    }
    __syncthreads();
  }

  // Store E (de-swizzle; swizzled rows are not 16B-contiguous, so plain stores)
#pragma unroll
  for (int t = 0; t < 16; ++t) {
    const int idx = tid + t * 256;
    const int r = idx >> 6, cc = idx & 63;
    io[base + idx] = Hc[HIDX(r, cc)];
  }
}

extern "C" void kernel_launch(void* const* d_in, const int* in_sizes, int n_in,
                              void* d_out, int out_size, void* d_ws, size_t ws_size,
                              hipStream_t stream) {
  (void)in_sizes; (void)n_in; (void)out_size; (void)d_ws; (void)ws_size;
  const float* dX  = (const float*)d_in[0];
  const float* Wre = (const float*)d_in[1];
  const float* Wim = (const float*)d_in[2];
  float2* out = (float2*)d_out;

  // Stage 1: AdX into d_out (complex64-interleaved), bandwidth bound (~256 MB out)
  adx_gemm_kernel<<<dim3(PN / 32, (PM * PM) / 64, PC), dim3(256), 0, stream>>>(
      dX, Wre, Wim, out);
  // Stage 2: in-place batched expm, fp32-WMMA compute bound (~400 GFLOP)
  expm64_kernel<<<dim3(PN * PC), dim3(256), 0, stream>>>(out);
}